// SSIM_5987184411024
// MI455X (gfx1250) — compile-verified
//
#include <hip/hip_runtime.h>

typedef float v2f __attribute__((ext_vector_type(2)));
typedef float v8f __attribute__((ext_vector_type(8)));

#define IMG_W   640
#define IMG_H   480
#define TW      64          // output tile cols per workgroup
#define TH      32          // output tile rows per workgroup
#define HALO    3
#define RAW_W   72          // 70 used, padded stride
#define RAW_H   38          // TH + 6
#define HROWS   38
#define HSTR    80          // 16 mod 64 -> conflict-free split-lane reads
#define C1_     (0.0001f)   // 0.01^2
#define C2_     (0.0009f)   // 0.03^2

// 7-tap Gaussian, sigma = 1.5, normalized (matches reference to fp32 accuracy)
__device__ __constant__ float G7[7] = {
    0.03663285f, 0.11128100f, 0.21674500f, 0.27068240f,
    0.21674500f, 0.11128100f, 0.03663285f};

__device__ __forceinline__ float clip01(float x) {
    return fminf(fmaxf(x, 0.001f), 0.999f);
}

__device__ __forceinline__ v8f wmma_f32x4(v2f a, v2f b, v8f c) {
#if __has_builtin(__builtin_amdgcn_wmma_f32_16x16x4_f32)
    // 8-arg pattern per probe: (neg_a, A, neg_b, B, c_mod, C, reuse_a, reuse_b)
    return __builtin_amdgcn_wmma_f32_16x16x4_f32(
        false, a, false, b, (short)0, c, false, false);
#else
    v8f d = c;
    asm volatile("v_wmma_f32_16x16x4_f32 %0, %1, %2, %0"
                 : "+v"(d) : "v"(a), "v"(b));
    return d;
#endif
}

__launch_bounds__(256)
__global__ void ssim_fused(const float* __restrict__ pred,
                           const float* __restrict__ targ,
                           double* __restrict__ acc) {
    __shared__ float Praw[RAW_H * RAW_W];
    __shared__ float Traw[RAW_H * RAW_W];
    __shared__ float Hpl[5][HROWS * HSTR];   // p, t, pp, tt, pt (horizontal blur)
    __shared__ float waveSums[8];

    const int tid = threadIdx.x;
    const int C0  = blockIdx.x * TW;
    const int R0  = blockIdx.y * TH;
    const size_t ibase = (size_t)blockIdx.z * ((size_t)IMG_W * IMG_H);

    // ---- Phase A: stage clipped inputs into LDS (zero = conv zero-padding) ----
    for (int idx = tid; idx < RAW_H * 70; idx += 256) {
        int r = idx / 70, c = idx % 70;
        int gr = R0 - HALO + r, gc = C0 - HALO + c;
        float pv = 0.f, tv = 0.f;
        if (gr >= 0 && gr < IMG_H && gc >= 0 && gc < IMG_W) {
            size_t o = ibase + (size_t)gr * IMG_W + gc;
            pv = clip01(pred[o]);
            tv = clip01(targ[o]);
        }
        Praw[r * RAW_W + c] = pv;
        Traw[r * RAW_W + c] = tv;
    }
    __syncthreads();

    // ---- Phase B: horizontal 7-tap Gaussian for the 5 maps (VALU) ----
    for (int idx = tid; idx < HROWS * TW; idx += 256) {
        int r = idx / TW, c = idx % TW;
        const float* pr = &Praw[r * RAW_W + c];
        const float* tr = &Traw[r * RAW_W + c];
        float hp = 0.f, ht = 0.f, hpp = 0.f, htt = 0.f, hpt = 0.f;
#pragma unroll
        for (int j = 0; j < 7; ++j) {
            float w  = G7[j];
            float pv = pr[j], tv = tr[j];
            hp  += w * pv;
            ht  += w * tv;
            hpp += w * (pv * pv);
            htt += w * (tv * tv);
            hpt += w * (pv * tv);
        }
        int o = r * HSTR + c;
        Hpl[0][o] = hp;  Hpl[1][o] = ht;  Hpl[2][o] = hpp;
        Hpl[3][o] = htt; Hpl[4][o] = hpt;
    }
    __syncthreads();

    // ---- Phase C: vertical pass as banded 16x24 x 24x16 matmul on WMMA ----
    const int lane = tid & 31;
    const int wave = tid >> 5;
    const int trb  = (wave >> 2) * 16;   // 2 row-blocks
    const int tcb  = (wave & 3) * 16;    // 4 col-blocks
    const int mrow = lane & 15;          // A: M index; B/C: N index
    const int koff = (lane < 16) ? 0 : 2;

    // A fragments: Wv[m][k] = G7[k-m] for 0<=k-m<=6, else 0 (k in 0..23)
    v2f aF[6];
#pragma unroll
    for (int kb = 0; kb < 6; ++kb) {
        int k0 = kb * 4 + koff;
        int d0 = k0 - mrow, d1 = d0 + 1;
        aF[kb].x = (d0 >= 0 && d0 < 7) ? G7[d0] : 0.f;
        aF[kb].y = (d1 >= 0 && d1 < 7) ? G7[d1] : 0.f;
    }

    v8f aP  = {}; v8f aT  = {}; v8f aPP = {}; v8f aTT = {}; v8f aPT = {};
#pragma unroll
    for (int kb = 0; kb < 6; ++kb) {
        int k0 = kb * 4 + koff;
        int k1 = k0 + 1;
        // clamp: k=22,23 have A==0, values irrelevant (but must be in-range/finite)
        int r0 = trb + ((k0 < 22) ? k0 : 21);
        int r1 = trb + ((k1 < 22) ? k1 : 21);
        int o0 = r0 * HSTR + tcb + mrow;
        int o1 = r1 * HSTR + tcb + mrow;
        v2f b;
        b.x = Hpl[0][o0]; b.y = Hpl[0][o1];  aP  = wmma_f32x4(aF[kb], b, aP);
        b.x = Hpl[1][o0]; b.y = Hpl[1][o1];  aT  = wmma_f32x4(aF[kb], b, aT);
        b.x = Hpl[2][o0]; b.y = Hpl[2][o1];  aPP = wmma_f32x4(aF[kb], b, aPP);
        b.x = Hpl[3][o0]; b.y = Hpl[3][o1];  aTT = wmma_f32x4(aF[kb], b, aTT);
        b.x = Hpl[4][o0]; b.y = Hpl[4][o1];  aPT = wmma_f32x4(aF[kb], b, aPT);
    }

    // ---- SSIM map + partial sum (each lane owns 8 pixels of its 16x16 tile) ----
    float lsum = 0.f;
#pragma unroll
    for (int i = 0; i < 8; ++i) {
        float mu1 = aP[i], mu2 = aT[i];
        float mu1s = mu1 * mu1;
        float mu2s = mu2 * mu2;
        float m12  = mu1 * mu2;
        float s1   = aPP[i] - mu1s;
        float s2   = aTT[i] - mu2s;
        float s12  = aPT[i] - m12;
        float num  = (2.f * m12 + C1_) * (2.f * s12 + C2_);
        float den  = (mu1s + mu2s + C1_) * (s1 + s2 + C2_);
        lsum += num / (den + 1e-8f);
    }

    // wave reduce (wave32)
#pragma unroll
    for (int off = 16; off > 0; off >>= 1)
        lsum += __shfl_xor(lsum, off, 32);
    if (lane == 0) waveSums[wave] = lsum;
    __syncthreads();
    if (tid == 0) {
        float s = 0.f;
#pragma unroll
        for (int w = 0; w < 8; ++w) s += waveSums[w];
        atomicAdd(acc, (double)s);   // global_atomic_add_f64
    }
}

__global__ void ssim_zero(double* acc) { acc[0] = 0.0; }

__global__ void ssim_finalize(const double* __restrict__ acc,
                              float* __restrict__ out, double invN) {
    out[0] = (float)(1.0 - acc[0] * invN);
}

extern "C" void kernel_launch(void* const* d_in, const int* in_sizes, int n_in,
                              void* d_out, int out_size, void* d_ws, size_t ws_size,
                              hipStream_t stream) {
    const float* pred = (const float*)d_in[0];
    const float* targ = (const float*)d_in[1];
    float* out  = (float*)d_out;
    double* acc = (double*)d_ws;

    const int plane = IMG_W * IMG_H;
    const int nimg  = in_sizes[0] / plane;           // 64
    const double invN = 1.0 / ((double)plane * (double)nimg);

    hipLaunchKernelGGL(ssim_zero, dim3(1), dim3(1), 0, stream, acc);
    dim3 grid(IMG_W / TW, IMG_H / TH, nimg);         // 10 x 15 x 64
    hipLaunchKernelGGL(ssim_fused, grid, dim3(256), 0, stream, pred, targ, acc);
    hipLaunchKernelGGL(ssim_finalize, dim3(1), dim3(1), 0, stream, acc, out, invN);
}